// InterHT_19370302505589
// MI455X (gfx1250) — compile-verified
//
#include <hip/hip_runtime.h>
#include <math.h>

#define EMB   512
#define WAVE  32
#define WPB   8                 // waves (rows) per block
#define TPB   (WPB * WAVE)      // 256 threads

typedef float v4f __attribute__((ext_vector_type(4)));

// wave32 xor-butterfly sum via ds_swizzle group-of-32 mode:
// offset = (xor_mask << 10) | (and_mask = 0x1f). All lanes end with the sum.
__device__ __forceinline__ float wave_sum(float v) {
  v += __int_as_float(__builtin_amdgcn_ds_swizzle(__float_as_int(v), 0x041f)); // xor 1
  v += __int_as_float(__builtin_amdgcn_ds_swizzle(__float_as_int(v), 0x081f)); // xor 2
  v += __int_as_float(__builtin_amdgcn_ds_swizzle(__float_as_int(v), 0x101f)); // xor 4
  v += __int_as_float(__builtin_amdgcn_ds_swizzle(__float_as_int(v), 0x201f)); // xor 8
  v += __int_as_float(__builtin_amdgcn_ds_swizzle(__float_as_int(v), 0x401f)); // xor 16
  return v;
}

__global__ __launch_bounds__(TPB) void interht_score_kernel(
    const float* __restrict__ head, const float* __restrict__ tail,
    const float* __restrict__ rel_tab, const int* __restrict__ rel_id,
    float* __restrict__ out, int rows)
{
  __shared__ float relbuf[WPB][EMB];          // 16 KB / block, wave-private slices
  const int wid  = (int)(threadIdx.x >> 5);
  const int lane = (int)(threadIdx.x & 31u);
  const int row  = (int)blockIdx.x * WPB + wid;
  if (row >= rows) return;

  // (a) Issue the relation-id gather first (1 load, retires before the stream).
  const int rid = rel_id[row];

  // (b) Stream head/tail with non-temporal b128 loads — the bulk 1.07 GB of
  //     one-touch traffic, kept out of L2 so the 2 MB relation table stays hot.
  const float* hrow = head + (size_t)row * (2 * EMB);
  const float* trow = tail + (size_t)row * (2 * EMB);
  const int c = lane * 16;

  v4f hm[4], ha[4], tm[4], ta[4];
#pragma unroll
  for (int i = 0; i < 4; ++i) {
    hm[i] = __builtin_nontemporal_load((const v4f*)(hrow +       c + 4 * i));
    ha[i] = __builtin_nontemporal_load((const v4f*)(hrow + EMB + c + 4 * i));
    tm[i] = __builtin_nontemporal_load((const v4f*)(trow +       c + 4 * i));
    ta[i] = __builtin_nontemporal_load((const v4f*)(trow + EMB + c + 4 * i));
  }

  // (c) Kick off async DMA of this wave's relation row into LDS (ASYNCcnt).
  //     GVS addressing: saddr = uniform row base (readfirstlane'd rid), vaddr =
  //     lane*16B; four b128 ops mirror the 2048-byte row into LDS. Overlaps
  //     with the norm pass below; drained only after it.
  {
    const float* relrow =
        rel_tab + (size_t)__builtin_amdgcn_readfirstlane(rid) * EMB;
    unsigned voff  = (unsigned)lane * 16u;
    unsigned lds_a = (unsigned)(size_t)&relbuf[wid][0] + voff;
    asm volatile(
        "global_load_async_to_lds_b128 %0, %1, %2 offset:0\n\t"
        "global_load_async_to_lds_b128 %0, %1, %2 offset:512\n\t"
        "global_load_async_to_lds_b128 %0, %1, %2 offset:1024\n\t"
        "global_load_async_to_lds_b128 %0, %1, %2 offset:1536"
        :: "v"(lds_a), "v"(voff), "s"(relrow)
        : "memory");
  }

  // ---- Norm pass (overlaps with the async relation DMA in flight). ----
  float s_hm = 0.f, s_ha = 0.f, s_tm = 0.f, s_ta = 0.f;
#pragma unroll
  for (int i = 0; i < 4; ++i)
#pragma unroll
    for (int j = 0; j < 4; ++j) {
      s_hm = fmaf(hm[i][j], hm[i][j], s_hm);
      s_ha = fmaf(ha[i][j], ha[i][j], s_ha);
      s_tm = fmaf(tm[i][j], tm[i][j], s_tm);
      s_ta = fmaf(ta[i][j], ta[i][j], s_ta);
    }
  s_hm = wave_sum(s_hm); s_ha = wave_sum(s_ha);
  s_tm = wave_sum(s_tm); s_ta = wave_sum(s_ta);

  const float i_hm = 1.0f / fmaxf(sqrtf(s_hm), 1e-12f);
  const float i_ha = 1.0f / fmaxf(sqrtf(s_ha), 1e-12f);
  const float i_tm = 1.0f / fmaxf(sqrtf(s_tm), 1e-12f);
  const float i_ta = 1.0f / fmaxf(sqrtf(s_ta), 1e-12f);

  // ---- Drain the relation DMA, then the distance pass from regs + LDS. ----
  asm volatile("s_wait_asynccnt 0" ::: "memory");

  float acc = 0.f;
#pragma unroll
  for (int i = 0; i < 4; ++i) {
    v4f rl = *(const v4f*)&relbuf[wid][c + 4 * i];
#pragma unroll
    for (int j = 0; j < 4; ++j) {
      float d = fmaf(hm[i][j] * i_hm, fmaf(ta[i][j], i_ta, 1.0f), rl[j])
              - tm[i][j] * i_tm * fmaf(ha[i][j], i_ha, 1.0f);
      acc = fmaf(d, d, acc);
    }
  }
  acc = wave_sum(acc);
  if (lane == 0) out[row] = -sqrtf(acc);
}

extern "C" void kernel_launch(void* const* d_in, const int* in_sizes, int n_in,
                              void* d_out, int out_size, void* d_ws, size_t ws_size,
                              hipStream_t stream) {
  (void)n_in; (void)d_ws; (void)ws_size; (void)out_size;
  const float* head    = (const float*)d_in[0];   // (B, 1024) f32
  const float* tail    = (const float*)d_in[1];   // (B, 1024) f32
  const float* rel_tab = (const float*)d_in[2];   // (1000, 512) f32
  const int*   rel_id  = (const int*)d_in[3];     // (B,) i32
  float* out = (float*)d_out;

  const int rows = in_sizes[3];                   // BATCH
  const int blocks = (rows + WPB - 1) / WPB;
  hipLaunchKernelGGL(interht_score_kernel, dim3(blocks), dim3(TPB), 0, stream,
                     head, tail, rel_tab, rel_id, out, rows);
}